// Gaussian_attn_70068096467941
// MI455X (gfx1250) — compile-verified
//
#include <hip/hip_runtime.h>

// Problem constants (match reference)
#define BB  8
#define SQb 512
#define SKb 512
#define DDp 32
#define COV (DDp * DDp)     // 1024
#define KX  2112            // 1024 (cov+outer) + 1024 (inv) + 32 + 32

typedef __attribute__((ext_vector_type(2))) float v2f;
typedef __attribute__((ext_vector_type(8))) float v8f;
typedef int v4i_ __attribute__((vector_size(16)));  // matches builtin param

__device__ __forceinline__ v8f wmma_f32_16x16x4(v2f a, v2f b, v8f c) {
  // D = A(16x4 f32) * B(4x16 f32) + C(16x16 f32)
  return __builtin_amdgcn_wmma_f32_16x16x4_f32(false, a, false, b, (short)0, c,
                                               false, false);
}

// CDNA5 async global->LDS staging (ASYNCcnt path), guarded so the file still
// compiles if this toolchain doesn't expose the builtins.
// Signature (leaked by the round-2 diagnostic): param0 is a non-const
// addrspace(1) pointer to vector_size(16) int; we match it exactly.
#if defined(__HIP_DEVICE_COMPILE__) &&                              \
    __has_builtin(__builtin_amdgcn_global_load_async_to_lds_b128) && \
    __has_builtin(__builtin_amdgcn_s_wait_asynccnt)
#define USE_ASYNC 1
#else
#define USE_ASYNC 0
#endif

#if USE_ASYNC
#define ASYNC_B128(gsrc, ldst)                                                \
  __builtin_amdgcn_global_load_async_to_lds_b128(                             \
      (__attribute__((address_space(1))) v4i_*)(unsigned long long)(gsrc),    \
      (__attribute__((address_space(3))) v4i_*)(ldst), 0, 0)
#define ASYNC_WAIT() __builtin_amdgcn_s_wait_asynccnt(0)
#endif

// ---------------------------------------------------------------------------
// Kernel 1: per-token prep. One wave32 per token. Lane owns column `lane` of
// the 32x32 covariance. Gauss-Jordan inverse fully in registers.
//   q side (isK=0): Xrow = [cov+m m^T | inv | m  | inv*m ]
//   k side (isK=1): Xrow = [inv | cov+m m^T | -2*inv*m | -2*m ],  sk = m^T inv m
// ---------------------------------------------------------------------------
__global__ __launch_bounds__(256) void prep_kernel(
    const float* __restrict__ mean, const float* __restrict__ cov,
    float* __restrict__ X, float* __restrict__ skout, int isK) {
  const int lane = threadIdx.x & 31;
  const int token = blockIdx.x * 8 + (threadIdx.x >> 5);  // 0..4095

  const float m = mean[(size_t)token * DDp + lane];
  float a[32], inv[32], mi[32];
#pragma unroll
  for (int i = 0; i < 32; ++i)
    a[i] = cov[(size_t)token * COV + i * DDp + lane];  // a[i] = A[i][lane]
#pragma unroll
  for (int i = 0; i < 32; ++i) mi[i] = __shfl(m, i, 32);  // broadcast m[i]

  float* Xrow = X + (size_t)token * KX;
  const int offCO = isK ? 1024 : 0;
  const int offIN = isK ? 0 : 1024;
#pragma unroll
  for (int i = 0; i < 32; ++i)
    Xrow[offCO + i * DDp + lane] = a[i] + mi[i] * m;  // cov + m m^T

  // Gauss-Jordan (SPD, well-conditioned, no pivoting). Lane c owns column c.
#pragma unroll
  for (int i = 0; i < 32; ++i) inv[i] = (i == lane) ? 1.0f : 0.0f;
#pragma unroll
  for (int kk = 0; kk < 32; ++kk) {
    const float p = __shfl(a[kk], kk, 32);  // pivot A[kk][kk]
    const float rp = 1.0f / p;
    a[kk] *= rp;
    inv[kk] *= rp;
#pragma unroll
    for (int i = 0; i < 32; ++i) {
      if (i == kk) continue;
      const float f = __shfl(a[i], kk, 32);  // A[i][kk] (column kk, lane kk)
      a[i] -= f * a[kk];
      inv[i] -= f * inv[kk];
    }
  }
#pragma unroll
  for (int i = 0; i < 32; ++i) Xrow[offIN + i * DDp + lane] = inv[i];

  // cv[lane] = (inv * m)[lane]   (inv is symmetric)
  float cv = 0.0f;
#pragma unroll
  for (int i = 0; i < 32; ++i) cv += inv[i] * mi[i];

  if (isK) {
    Xrow[2048 + lane] = -2.0f * cv;
    Xrow[2080 + lane] = -2.0f * m;
    float s = m * cv;  // m^T inv m
#pragma unroll
    for (int off = 16; off >= 1; off >>= 1) s += __shfl_xor(s, off, 32);
    if (lane == 0) skout[token] = s;
  } else {
    Xrow[2048 + lane] = m;
    Xrow[2080 + lane] = cv;
  }
}

// ---------------------------------------------------------------------------
// WMMA GEMM tiling shared by the two big GEMMs:
// block = 256 threads = 8 waves arranged 4(M) x 2(N); block tile 128 x 64;
// each wave computes a 32x32 tile = 2x2 WMMA accumulators; K step = 16.
// A (and logits-B) LDS rows padded to stride 20 floats (80B: 16B-aligned rows,
// conflict-free strided float2 fragment loads). Double-buffered with async
// global->LDS loads when available (one barrier per K step).
// ---------------------------------------------------------------------------
#define TSTR 20

// Kernel 2: logits S[b,q,k] = 0.25*( Xq[b,q,:] . Yk[b,k,:] + sk[b,k] )
__global__ __launch_bounds__(256) void logits_kernel(
    const float* __restrict__ Xq, const float* __restrict__ Yk,
    const float* __restrict__ sk, float* __restrict__ S) {
  __shared__ __align__(16) float As[2][128 * TSTR];
  __shared__ __align__(16) float Bs[2][64 * TSTR];
  const int tid = threadIdx.x;
  const int lane = tid & 31, l16 = lane & 15, h = lane >> 4;
  const int w = tid >> 5, wM = w & 3, wN = w >> 2;
  const int b = blockIdx.z;
  const int q0 = blockIdx.x * 128;
  const int k0 = blockIdx.y * 64;

  const float* Arow = Xq + ((size_t)b * SQb + q0) * KX;  // [128][KX], M-major
  const float* Brow = Yk + ((size_t)b * SKb + k0) * KX;  // [64][KX],  N-major

  // staging indices: one float4 of B + two float4 of A per thread per chunk
  const int ar = tid >> 2, ac = (tid & 3) * 4;  // A rows ar, ar+64; col ac
  v8f acc[2][2] = {};

  auto compute = [&](const float* Ab, const float* Bb) {
#pragma unroll
    for (int s = 0; s < 4; ++s) {
      const int ko = s * 4 + 2 * h;  // lane's K pair within the 16-chunk
      v2f a0 = *(const v2f*)&Ab[(wM * 32 + l16) * TSTR + ko];
      v2f a1 = *(const v2f*)&Ab[(wM * 32 + 16 + l16) * TSTR + ko];
      v2f b0 = *(const v2f*)&Bb[(wN * 32 + l16) * TSTR + ko];
      v2f b1 = *(const v2f*)&Bb[(wN * 32 + 16 + l16) * TSTR + ko];
      acc[0][0] = wmma_f32_16x16x4(a0, b0, acc[0][0]);
      acc[0][1] = wmma_f32_16x16x4(a0, b1, acc[0][1]);
      acc[1][0] = wmma_f32_16x16x4(a1, b0, acc[1][0]);
      acc[1][1] = wmma_f32_16x16x4(a1, b1, acc[1][1]);
    }
  };

#if USE_ASYNC
  ASYNC_B128(Arow + (size_t)ar * KX + ac, &As[0][ar * TSTR + ac]);
  ASYNC_B128(Arow + (size_t)(ar + 64) * KX + ac, &As[0][(ar + 64) * TSTR + ac]);
  ASYNC_B128(Brow + (size_t)ar * KX + ac, &Bs[0][ar * TSTR + ac]);
  for (int kc = 0; kc < KX / 16; ++kc) {
    const int cur = kc & 1;
    ASYNC_WAIT();        // this wave's loads into buffer `cur` are done
    __syncthreads();     // ... and so are everyone else's; `cur^1` reads done
    if (kc + 1 < KX / 16) {
      const int nxt = cur ^ 1;
      const int kb = (kc + 1) * 16;
      ASYNC_B128(Arow + (size_t)ar * KX + kb + ac, &As[nxt][ar * TSTR + ac]);
      ASYNC_B128(Arow + (size_t)(ar + 64) * KX + kb + ac,
                 &As[nxt][(ar + 64) * TSTR + ac]);
      ASYNC_B128(Brow + (size_t)ar * KX + kb + ac, &Bs[nxt][ar * TSTR + ac]);
    }
    compute(As[cur], Bs[cur]);
  }
#else
  for (int kc = 0; kc < KX / 16; ++kc) {
    const float4 va0 = *(const float4*)(Arow + (size_t)ar * KX + kc * 16 + ac);
    const float4 va1 =
        *(const float4*)(Arow + (size_t)(ar + 64) * KX + kc * 16 + ac);
    const float4 vb = *(const float4*)(Brow + (size_t)ar * KX + kc * 16 + ac);
    *(float4*)&As[0][ar * TSTR + ac] = va0;
    *(float4*)&As[0][(ar + 64) * TSTR + ac] = va1;
    *(float4*)&Bs[0][ar * TSTR + ac] = vb;
    __syncthreads();
    compute(As[0], Bs[0]);
    __syncthreads();
  }
#endif

#pragma unroll
  for (int tM = 0; tM < 2; ++tM)
#pragma unroll
    for (int tN = 0; tN < 2; ++tN) {
      const int col = k0 + wN * 32 + tN * 16 + l16;
      const float skv = sk[b * SKb + col];
#pragma unroll
      for (int r = 0; r < 8; ++r) {
        const int row = q0 + wM * 32 + tM * 16 + r + 8 * h;
        S[((size_t)b * SQb + row) * SKb + col] = 0.25f * (acc[tM][tN][r] + skv);
      }
    }
}

// Kernel 3: row softmax over k (512), one wave per row.
__global__ __launch_bounds__(256) void softmax_kernel(float* __restrict__ S) {
  const int lane = threadIdx.x & 31;
  const int row = blockIdx.x * 8 + (threadIdx.x >> 5);
  float* p = S + (size_t)row * SKb;
  float v[16];
  float mx = -3.4e38f;
#pragma unroll
  for (int j = 0; j < 16; ++j) {
    v[j] = p[j * 32 + lane];
    mx = fmaxf(mx, v[j]);
  }
#pragma unroll
  for (int off = 16; off >= 1; off >>= 1) mx = fmaxf(mx, __shfl_xor(mx, off, 32));
  float sum = 0.0f;
#pragma unroll
  for (int j = 0; j < 16; ++j) {
    v[j] = __expf(v[j] - mx);
    sum += v[j];
  }
#pragma unroll
  for (int off = 16; off >= 1; off >>= 1) sum += __shfl_xor(sum, off, 32);
  const float rs = 1.0f / sum;
#pragma unroll
  for (int j = 0; j < 16; ++j) p[j * 32 + lane] = v[j] * rs;
}

// Kernel 4: out_cov[b,q,:] = S[b,q,:] @ v_cov[b,:,:1024]   (WMMA GEMM)
// B chunk staged untransposed as [16 K][64 N] (async-B128 friendly); B
// fragments are two b32 LDS loads (lanes read consecutive N -> no conflicts).
__global__ __launch_bounds__(256) void outcov_kernel(
    const float* __restrict__ S, const float* __restrict__ vcov,
    float* __restrict__ outcov) {
  __shared__ __align__(16) float As[2][128 * TSTR];
  __shared__ __align__(16) float Bs[2][16 * 64];
  const int tid = threadIdx.x;
  const int lane = tid & 31, l16 = lane & 15, h = lane >> 4;
  const int w = tid >> 5, wM = w & 3, wN = w >> 2;
  const int b = blockIdx.z;
  const int q0 = blockIdx.x * 128;
  const int n0 = blockIdx.y * 64;

  const float* Arow = S + ((size_t)b * SQb + q0) * SKb;  // [128][512]
  const float* Bbase = vcov + (size_t)b * SKb * COV;     // [512][1024], K-major

  const int ar = tid >> 2, ac = (tid & 3) * 4;           // A staging
  const int bk = tid >> 4, bn = (tid & 15) * 4;          // B staging (16x64)
  v8f acc[2][2] = {};

  auto compute = [&](const float* Ab, const float* Bb) {
#pragma unroll
    for (int s = 0; s < 4; ++s) {
      const int ko = s * 4 + 2 * h;
      v2f a0 = *(const v2f*)&Ab[(wM * 32 + l16) * TSTR + ko];
      v2f a1 = *(const v2f*)&Ab[(wM * 32 + 16 + l16) * TSTR + ko];
      const int nb0 = wN * 32 + l16, nb1 = wN * 32 + 16 + l16;
      v2f b0 = {Bb[(ko + 0) * 64 + nb0], Bb[(ko + 1) * 64 + nb0]};
      v2f b1 = {Bb[(ko + 0) * 64 + nb1], Bb[(ko + 1) * 64 + nb1]};
      acc[0][0] = wmma_f32_16x16x4(a0, b0, acc[0][0]);
      acc[0][1] = wmma_f32_16x16x4(a0, b1, acc[0][1]);
      acc[1][0] = wmma_f32_16x16x4(a1, b0, acc[1][0]);
      acc[1][1] = wmma_f32_16x16x4(a1, b1, acc[1][1]);
    }
  };

#if USE_ASYNC
  ASYNC_B128(Arow + (size_t)ar * SKb + ac, &As[0][ar * TSTR + ac]);
  ASYNC_B128(Arow + (size_t)(ar + 64) * SKb + ac, &As[0][(ar + 64) * TSTR + ac]);
  ASYNC_B128(Bbase + (size_t)bk * COV + n0 + bn, &Bs[0][bk * 64 + bn]);
  for (int kc = 0; kc < SKb / 16; ++kc) {
    const int cur = kc & 1;
    ASYNC_WAIT();
    __syncthreads();
    if (kc + 1 < SKb / 16) {
      const int nxt = cur ^ 1;
      const int kb = (kc + 1) * 16;
      ASYNC_B128(Arow + (size_t)ar * SKb + kb + ac, &As[nxt][ar * TSTR + ac]);
      ASYNC_B128(Arow + (size_t)(ar + 64) * SKb + kb + ac,
                 &As[nxt][(ar + 64) * TSTR + ac]);
      ASYNC_B128(Bbase + (size_t)(kb + bk) * COV + n0 + bn,
                 &Bs[nxt][bk * 64 + bn]);
    }
    compute(As[cur], Bs[cur]);
  }
#else
  for (int kc = 0; kc < SKb / 16; ++kc) {
    const float4 va0 = *(const float4*)(Arow + (size_t)ar * SKb + kc * 16 + ac);
    const float4 va1 =
        *(const float4*)(Arow + (size_t)(ar + 64) * SKb + kc * 16 + ac);
    const float4 vb =
        *(const float4*)(Bbase + (size_t)(kc * 16 + bk) * COV + n0 + bn);
    *(float4*)&As[0][ar * TSTR + ac] = va0;
    *(float4*)&As[0][(ar + 64) * TSTR + ac] = va1;
    *(float4*)&Bs[0][bk * 64 + bn] = vb;
    __syncthreads();
    compute(As[0], Bs[0]);
    __syncthreads();
  }
#endif

#pragma unroll
  for (int tM = 0; tM < 2; ++tM)
#pragma unroll
    for (int tN = 0; tN < 2; ++tN) {
      const int col = n0 + wN * 32 + tN * 16 + l16;
#pragma unroll
      for (int r = 0; r < 8; ++r) {
        const int row = q0 + wM * 32 + tM * 16 + r + 8 * h;
        outcov[((size_t)b * SQb + row) * COV + col] = acc[tM][tN][r];
      }
    }
}

// Kernel 5: out_mean = S @ v_mean (tiny: 134 MFLOP, plain VALU)
__global__ __launch_bounds__(256) void outmean_kernel(
    const float* __restrict__ S, const float* __restrict__ vmean,
    float* __restrict__ outmean) {
  const int t = blockIdx.x * 256 + threadIdx.x;  // 0..131071
  const int d = t & 31;
  const int q = (t >> 5) & 511;
  const int b = t >> 14;
  const float* srow = S + ((size_t)b * SQb + q) * SKb;
  const float* vm = vmean + (size_t)b * SKb * DDp + d;
  float s = 0.0f;
#pragma unroll 4
  for (int k = 0; k < SKb; ++k) s += srow[k] * vm[(size_t)k * DDp];
  outmean[t] = s;
}

// ---------------------------------------------------------------------------
extern "C" void kernel_launch(void* const* d_in, const int* in_sizes, int n_in,
                              void* d_out, int out_size, void* d_ws,
                              size_t ws_size, hipStream_t stream) {
  (void)in_sizes; (void)n_in; (void)out_size; (void)ws_size;
  const float* q_mean = (const float*)d_in[0];
  const float* q_cov  = (const float*)d_in[1];
  const float* k_mean = (const float*)d_in[2];
  const float* k_cov  = (const float*)d_in[3];
  const float* v_mean = (const float*)d_in[4];
  const float* v_cov  = (const float*)d_in[5];

  float* out_mean = (float*)d_out;                        // [B,SQ,32]
  float* out_cov  = out_mean + (size_t)BB * SQb * DDp;    // [B,SQ,32,32]

  // Workspace layout (floats): Xq[B*SQ][2112], Yk[B*SK][2112], sk[B*SK],
  // S[B*SQ][SK]  -> ~77.6 MB total
  float* Xq = (float*)d_ws;
  float* Yk = Xq + (size_t)BB * SQb * KX;
  float* sk = Yk + (size_t)BB * SKb * KX;
  float* Sc = sk + (size_t)BB * SKb;

  prep_kernel<<<512, 256, 0, stream>>>(q_mean, q_cov, Xq, nullptr, 0);
  prep_kernel<<<512, 256, 0, stream>>>(k_mean, k_cov, Yk, sk, 1);
  logits_kernel<<<dim3(4, 8, 8), 256, 0, stream>>>(Xq, Yk, sk, Sc);
  softmax_kernel<<<512, 256, 0, stream>>>(Sc);
  outcov_kernel<<<dim3(4, 16, 8), 256, 0, stream>>>(Sc, v_cov, out_cov);
  outmean_kernel<<<512, 256, 0, stream>>>(Sc, v_mean, out_mean);
}